// SinkhornDistance_14705968022209
// MI455X (gfx1250) — compile-verified
//
#include <hip/hip_runtime.h>
#include <math.h>

#define BB 4
#define PP 2048
#define DD 64
#define EPSV 0.1f
#define INV_EPS 10.0f
#define NMAT (BB * PP * PP)  // 16,777,216 elements per matrix

typedef float v2f __attribute__((ext_vector_type(2)));
typedef float v8f __attribute__((ext_vector_type(8)));

__global__ void sk_zero_kernel(float* p, int n) {
    int i = blockIdx.x * blockDim.x + threadIdx.x;
    if (i < n) p[i] = 0.0f;
}

// C[b][i][j] = |x_i|^2 + |y_j|^2 - 2*dot(x_i, y_j); also writes CT[b][j][i].
// One wave per 16x16 tile, K-loop of 16 V_WMMA_F32_16X16X4_F32.
__global__ __launch_bounds__(256) void sk_cost_kernel(const float* __restrict__ x,
                                                      const float* __restrict__ y,
                                                      float* __restrict__ C,
                                                      float* __restrict__ CT) {
    const int lane = threadIdx.x & 31;
    const int tile = blockIdx.x * 8 + (threadIdx.x >> 5);  // 0..65535
    const int b  = tile >> 14;          // 128*128 tiles per batch
    const int mt = (tile >> 7) & 127;
    const int nt = tile & 127;
    const int h  = lane >> 4;           // lane half selects K pair
    const int l  = lane & 15;

    const float* xr = x + ((size_t)b * PP + (size_t)mt * 16 + l) * DD;
    const float* yr = y + ((size_t)b * PP + (size_t)nt * 16 + l) * DD;

    v8f acc = {0.f, 0.f, 0.f, 0.f, 0.f, 0.f, 0.f, 0.f};
    float x2p = 0.f, y2p = 0.f;
#pragma unroll
    for (int kk = 0; kk < DD; kk += 4) {
        // A 16x4 f32 layout: lanes0-15 K={kk,kk+1}, lanes16-31 K={kk+2,kk+3}
        v2f a = *(const v2f*)(xr + kk + 2 * h);
        v2f bv = *(const v2f*)(yr + kk + 2 * h);
        x2p += a.x * a.x + a.y * a.y;
        y2p += bv.x * bv.x + bv.y * bv.y;
        acc = __builtin_amdgcn_wmma_f32_16x16x4_f32(false, a, false, bv,
                                                    (short)0, acc, false, false);
    }
    // each lane saw half the K values for its row; partner lane (xor 16) has the rest
    float x2 = x2p + __shfl_xor(x2p, 16, 32);
    float y2 = y2p + __shfl_xor(y2p, 16, 32);  // y2 for n = lane&15 (what D needs)

    const size_t base = (size_t)b * PP * PP;
#pragma unroll
    for (int r = 0; r < 8; r++) {
        // D layout: VGPR r -> M = r (lanes 0-15) / 8+r (lanes 16-31), N = lane&15
        int m = r + 8 * h;
        float xm = __shfl(x2, m, 32);  // lane m holds |x_m|^2
        float val = xm + y2 - 2.0f * acc[r];
        int gm = mt * 16 + m, gn = nt * 16 + l;
        C[base + (size_t)gm * PP + gn] = val;
        CT[base + (size_t)gn * PP + gm] = val;
    }
}

// One block per output row. out[row] = EPS*log(1/P + 1e-8) - EPS*LSE_j((din[j]-M[row][j])/EPS)
__global__ __launch_bounds__(256) void sk_lse_pass(const float* __restrict__ M,
                                                   const float* __restrict__ din,
                                                   float* __restrict__ dout) {
    __shared__ float red[8];
    const int row = blockIdx.x;          // 0..B*P-1
    const int b = row >> 11;
    const float* Mr = M + (size_t)row * PP;
    const float* dv = din + (size_t)b * PP;

    float t[8];
#pragma unroll
    for (int e = 0; e < 8; e++) {
        int j = threadIdx.x + 256 * e;
        t[e] = (dv[j] - Mr[j]) * INV_EPS;
    }
    float m = t[0];
#pragma unroll
    for (int e = 1; e < 8; e++) m = fmaxf(m, t[e]);
#pragma unroll
    for (int off = 16; off > 0; off >>= 1) m = fmaxf(m, __shfl_xor(m, off, 32));
    const int wid = threadIdx.x >> 5;
    if ((threadIdx.x & 31) == 0) red[wid] = m;
    __syncthreads();
    float Mx = red[0];
#pragma unroll
    for (int k = 1; k < 8; k++) Mx = fmaxf(Mx, red[k]);
    __syncthreads();

    float s = 0.f;
#pragma unroll
    for (int e = 0; e < 8; e++) s += expf(t[e] - Mx);
#pragma unroll
    for (int off = 16; off > 0; off >>= 1) s += __shfl_xor(s, off, 32);
    if ((threadIdx.x & 31) == 0) red[wid] = s;
    __syncthreads();
    if (threadIdx.x == 0) {
        float S = red[0];
#pragma unroll
        for (int k = 1; k < 8; k++) S += red[k];
        const float logmu = logf(1.0f / (float)PP + 1e-8f);
        dout[row] = EPSV * logmu - EPSV * (Mx + logf(S));
    }
}

// pi = exp((-C + u_i + v_j)/EPS); per-row partial of sum(pi*C) to ws (deterministic).
__global__ __launch_bounds__(256) void sk_pi_kernel(const float* __restrict__ C,
                                                    const float* __restrict__ u,
                                                    const float* __restrict__ v,
                                                    float* __restrict__ pi,
                                                    float* __restrict__ partial) {
    __shared__ float red[8];
    const int row = blockIdx.x;
    const int b = row >> 11;
    const float* Cr = C + (size_t)row * PP;
    float* Pr = pi + (size_t)row * PP;
    const float ui = u[row];
    const float* vb = v + (size_t)b * PP;

    float loc = 0.f;
#pragma unroll
    for (int e = 0; e < 8; e++) {
        int j = threadIdx.x + 256 * e;
        float c = Cr[j];
        float p = expf((ui + vb[j] - c) * INV_EPS);
        Pr[j] = p;
        loc += p * c;
    }
#pragma unroll
    for (int off = 16; off > 0; off >>= 1) loc += __shfl_xor(loc, off, 32);
    if ((threadIdx.x & 31) == 0) red[threadIdx.x >> 5] = loc;
    __syncthreads();
    if (threadIdx.x == 0) {
        float S = red[0];
#pragma unroll
        for (int k = 1; k < 8; k++) S += red[k];
        partial[row] = S;
    }
}

__global__ __launch_bounds__(256) void sk_cost_reduce(const float* __restrict__ partial,
                                                      float* __restrict__ out) {
    __shared__ float red[8];
    const int b = blockIdx.x;
    float s = 0.f;
#pragma unroll
    for (int e = 0; e < 8; e++) s += partial[b * PP + threadIdx.x + 256 * e];
#pragma unroll
    for (int off = 16; off > 0; off >>= 1) s += __shfl_xor(s, off, 32);
    if ((threadIdx.x & 31) == 0) red[threadIdx.x >> 5] = s;
    __syncthreads();
    if (threadIdx.x == 0) {
        float S = red[0];
#pragma unroll
        for (int k = 1; k < 8; k++) S += red[k];
        out[b] = S;
    }
}

extern "C" void kernel_launch(void* const* d_in, const int* in_sizes, int n_in,
                              void* d_out, int out_size, void* d_ws, size_t ws_size,
                              hipStream_t stream) {
    (void)in_sizes; (void)n_in; (void)out_size; (void)ws_size;
    const float* x = (const float*)d_in[0];
    const float* y = (const float*)d_in[1];

    float* out  = (float*)d_out;
    float* cost = out;                       // [4]
    float* pi   = out + 4;                   // [B*P*P]
    float* C    = out + 4 + (size_t)NMAT;    // [B*P*P]
    float* CT   = pi;                        // scratch: C^T lives here until the end

    float* u       = (float*)d_ws;           // [B*P]
    float* v       = u + BB * PP;            // [B*P]
    float* partial = v + BB * PP;            // [B*P]

    sk_zero_kernel<<<(2 * BB * PP + 255) / 256, 256, 0, stream>>>(u, 2 * BB * PP);

    // 65536 16x16 tiles, 8 waves (tiles) per 256-thread block
    sk_cost_kernel<<<(BB * 128 * 128) / 8, 256, 0, stream>>>(x, y, C, CT);

    for (int it = 0; it < 100; it++) {
        sk_lse_pass<<<BB * PP, 256, 0, stream>>>(C,  v, u);   // u update (row LSE)
        sk_lse_pass<<<BB * PP, 256, 0, stream>>>(CT, u, v);   // v update (col LSE via C^T)
    }

    sk_pi_kernel<<<BB * PP, 256, 0, stream>>>(C, u, v, pi, partial);
    sk_cost_reduce<<<BB, 256, 0, stream>>>(partial, cost);
}